// LinearGaussian_65781719105918
// MI455X (gfx1250) — compile-verified
//
#include <hip/hip_runtime.h>

typedef float v2f __attribute__((ext_vector_type(2)));
typedef float v4f __attribute__((ext_vector_type(4)));
typedef float v8f __attribute__((ext_vector_type(8)));

#define B_   256
#define IN_  512
#define OUT_ 512

// ---------------------------------------------------------------------------
// Kernel 1: y_mean = x_mean @ A_mean + b_mean
//           t1     = (diag(x_var) + x_mean^2) @ A_var + b_var   (workspace)
// Tiny (0.27 GFLOP total); A_mean/A_var (1 MB each) stay hot in the 192 MB L2.
// ---------------------------------------------------------------------------
__global__ __launch_bounds__(256) void k_mean_t1(
    const float* __restrict__ x_mean, const float* __restrict__ x_var,
    const float* __restrict__ A_mean, const float* __restrict__ A_var,
    const float* __restrict__ b_mean, const float* __restrict__ b_var,
    float* __restrict__ y_mean, float* __restrict__ t1)
{
    const int o = blockIdx.x * 256 + threadIdx.x;   // 0..511
    const int b = blockIdx.y;                       // 0..255

    __shared__ float sxm[IN_];
    __shared__ float sxx[IN_];
    for (int i = threadIdx.x; i < IN_; i += 256) {
        float xm = x_mean[b * IN_ + i];
        sxm[i] = xm;
        // diagonal of x_var[b]
        sxx[i] = x_var[(size_t)b * IN_ * IN_ + (size_t)i * (IN_ + 1)] + xm * xm;
    }
    __syncthreads();

    float am = 0.0f, av = 0.0f;
    for (int i = 0; i < IN_; ++i) {
        am = fmaf(sxm[i], A_mean[i * OUT_ + o], am);
        av = fmaf(sxx[i], A_var [i * OUT_ + o], av);
    }
    y_mean[b * OUT_ + o] = am + b_mean[o];
    t1    [b * OUT_ + o] = av + b_var[o];
}

// ---------------------------------------------------------------------------
// Kernel 2: term2_diag via fp32 WMMA, plus the diag_embed write of y_var.
//   term2[b,o] = sum_i A_mean[i,o] * M[b,i,o],  M[b] = x_var[b] @ A_mean
// One workgroup = (batch b, 16-wide o-tile). 4 waves; wave w owns row-tiles
// [w*128, w*128+128) with 8 independent accumulators.
// ---------------------------------------------------------------------------
__global__ __launch_bounds__(128) void k_var(
    const float* __restrict__ x_var, const float* __restrict__ A_mean,
    const float* __restrict__ t1, float* __restrict__ y_var)
{
    const int ot  = blockIdx.x;          // 0..31 : o tile
    const int b   = blockIdx.y;          // 0..255
    const int tid = threadIdx.x;
    const int w   = tid >> 5;            // wave 0..3
    const int l   = tid & 31;            // lane
    const int lh  = l >> 4;              // lane half
    const int ln  = l & 15;              // lane-in-half

    // A_mean o-tile pre-packed as WMMA-B operands: [c][lane][2]
    //   AmPack[c*64 + lane*2 + k] = A_mean[4c + (lane>>4) + 2k][ot*16 + (lane&15)]
    // Inverse map for element (j, o):
    //   idx = (j>>2)*64 + ((j&1)*16 + o)*2 + ((j>>1)&1)
    __shared__ float AmPack[(IN_ / 4) * 64];        // 32 KB
    __shared__ float red[4 * 32];

    for (int idx = tid; idx < IN_ * 16; idx += 128) {
        int c    = idx >> 6;
        int r    = idx & 63;
        int lane = r >> 1;
        int k    = r & 1;
        int j    = 4 * c + (lane >> 4) + 2 * k;
        int o    = lane & 15;
        AmPack[idx] = A_mean[j * OUT_ + ot * 16 + o];
    }
    __syncthreads();

    // A-operand base: x_var[b, w*128 + t*16 + ln, 4c + 2*lh .. +1]
    const float* Xb   = x_var + (size_t)b * IN_ * IN_;
    const float* arow = Xb + (size_t)(w * 128 + ln) * IN_ + 2 * lh;

    v8f acc[8] = {};
    for (int c = 0; c < IN_ / 4; ++c) {
        v2f bop = *(const v2f*)&AmPack[c * 64 + l * 2];   // ds_load_b64, no bank conflict
#pragma unroll
        for (int t = 0; t < 8; ++t) {
            v2f a = *(const v2f*)(arow + (size_t)t * 16 * IN_ + c * 4);  // global b64
            acc[t] = __builtin_amdgcn_wmma_f32_16x16x4_f32(
                false, a, false, bop, (short)0, acc[t], false, false);
        }
    }

    // Elementwise reduce: s = sum over this lane's (i, o=ln) of A_mean[i,o]*M[i,o]
    // C-layout: acc[t][v] is (i = w*128 + t*16 + v + 8*lh, o = ln)
    float s = 0.0f;
#pragma unroll
    for (int t = 0; t < 8; ++t) {
        int ibase = w * 128 + t * 16 + 8 * lh;
#pragma unroll
        for (int v = 0; v < 8; ++v) {
            int j = ibase + v;
            float amc = AmPack[(j >> 2) * 64 + ((j & 1) * 16 + ln) * 2 + ((j >> 1) & 1)];
            s = fmaf(acc[t][v], amc, s);
        }
    }
    red[w * 32 + l] = s;
    __syncthreads();

    // Zero-fill this workgroup's 16 rows of y_var[b] (diag_embed off-diagonals).
    // Write-once stream (268 MB across the grid): non-temporal so it does not
    // evict the x_var read stream / A tiles from L2.
    float* yb = y_var + (size_t)b * OUT_ * OUT_ + (size_t)ot * 16 * OUT_;
    v4f z = {};
    for (int k = tid; k < 16 * OUT_ / 4; k += 128)
        __builtin_nontemporal_store(z, &((v4f*)yb)[k]);
    __syncthreads();

    // Diagonal: y_var[b, r, r] = term1+b_var (from ws) + term2
    if (tid < 16) {
        float t2 = red[0 * 32 + tid] + red[0 * 32 + tid + 16]
                 + red[1 * 32 + tid] + red[1 * 32 + tid + 16]
                 + red[2 * 32 + tid] + red[2 * 32 + tid + 16]
                 + red[3 * 32 + tid] + red[3 * 32 + tid + 16];
        int r = ot * 16 + tid;
        __builtin_nontemporal_store(t2 + t1[b * OUT_ + r], &yb[(size_t)tid * OUT_ + r]);
    }
}

// ---------------------------------------------------------------------------
extern "C" void kernel_launch(void* const* d_in, const int* in_sizes, int n_in,
                              void* d_out, int out_size, void* d_ws, size_t ws_size,
                              hipStream_t stream) {
    (void)in_sizes; (void)n_in; (void)out_size; (void)ws_size;
    const float* x_mean = (const float*)d_in[0];
    const float* x_var  = (const float*)d_in[1];
    const float* A_mean = (const float*)d_in[2];
    const float* A_var  = (const float*)d_in[3];
    const float* b_mean = (const float*)d_in[4];
    const float* b_var  = (const float*)d_in[5];

    float* y_mean = (float*)d_out;                          // [B, OUT]
    float* y_var  = (float*)d_out + (size_t)B_ * OUT_;      // [B, OUT, OUT]
    float* t1     = (float*)d_ws;                           // [B, OUT] scratch

    k_mean_t1<<<dim3(OUT_ / 256, B_), 256, 0, stream>>>(
        x_mean, x_var, A_mean, A_var, b_mean, b_var, y_mean, t1);

    k_var<<<dim3(32, B_), 128, 0, stream>>>(x_var, A_mean, t1, y_var);
}